// UntilLayer_57071525429384
// MI455X (gfx1250) — compile-verified
//
#include <hip/hip_runtime.h>
#include <hip/hip_bf16.h>

#define BATCH   64
#define NPATCH  64
#define BIN1    10
#define BIN_T   16
#define EPS2    0.001f
#define OUT_DIM (NPATCH * BIN1 * (BIN1 - 1))   // 5760
#define ROWLEN  (BIN_T * BIN1)                 // 160 floats per (B,p) slab

typedef __attribute__((ext_vector_type(2))) float v2f;
typedef __attribute__((ext_vector_type(8))) float v8f;

// Fast reciprocal: single v_rcp_f32 (1 ulp) instead of the full IEEE
// div_scale/div_fmas/div_fixup sequence.
__device__ __forceinline__ float fast_rcp(float x) {
    return __builtin_amdgcn_rcpf(x);
}

__device__ __forceinline__ float fast_sigmoid(float z) {
    return fast_rcp(1.0f + __expf(-z));   // v_exp_f32 + v_add + v_rcp
}

// One block per (p, j, m); 64 threads = 64 batches.
__global__ __launch_bounds__(64) void until_layer_kernel(
    const float* __restrict__ x,        // (64, 64, 16, 10) = (B, p, v, bin)
    const float* __restrict__ bound1,   // (5760,) indexed by p*90 + j*9 + m
    const float* __restrict__ bound2,   // (5760,)
    float* __restrict__ out)            // (64, 5760)
{
    __shared__ __align__(16) float W1s[16][16];  // W1s[u][v]; rows 64B apart
    __shared__ float Wfs[16];

    const int idx = blockIdx.x;          // p*90 + j*9 + m
    const int m   = idx % (BIN1 - 1);
    const int j   = (idx / (BIN1 - 1)) % BIN1;
    const int p   = idx / (BIN1 * (BIN1 - 1));
    const int k   = (m < j) ? m : (m + 1);     // kidx[j][m]

    const float b1 = bound1[idx];
    const float b2 = bound2[idx];
    const float Ws = fminf(b1, b2);            // W_start
    const float We = fmaxf(b1, b2);            // W_end

    const int tid = threadIdx.x;

    if (tid < 32) {
        // Wave 0: z[u][v] = (v - Ws) * (u - EPS2 - v), rank-2 -> one WMMA.
        //   A (16x4):  A[u,0]=u, A[u,1]=1, A[u,2..3]=0
        //   Bm (4x16): Bm[0,v]=(v-Ws), Bm[1,v]=-(EPS2+v)*(v-Ws), rows 2..3 = 0
        // f32 16x4 A layout: lanes 0-15 -> M=lane, VGPR0=K0, VGPR1=K1;
        //                    lanes 16-31 -> M=lane-16, VGPR0=K2, VGPR1=K3.
        const int  lane = tid;
        const bool hi   = lane >= 16;
        v2f av, bv;
        if (!hi) {
            const float fl = (float)lane;
            av.x = fl;            // A[u=lane, k=0]
            av.y = 1.0f;          // A[u=lane, k=1]
            bv.x = fl - Ws;                      // Bm[0, v=lane]
            bv.y = -(EPS2 + fl) * (fl - Ws);     // Bm[1, v=lane]
        } else {
            av.x = 0.0f; av.y = 0.0f;            // K=2,3 padding
            bv.x = 0.0f; bv.y = 0.0f;
        }
        v8f c = {};
        // (neg_a, A, neg_b, B, c_mod, C, reuse_a, reuse_b)
        v8f d = __builtin_amdgcn_wmma_f32_16x16x4_f32(
            false, av, false, bv, (short)0, c, false, false);

        // C/D layout: VGPR r holds z[M = r + 8*hi][N = lane & 15]
        const int n  = lane & 15;
        const int mb = hi ? 8 : 0;
        #pragma unroll
        for (int r = 0; r < 8; ++r) {
            W1s[r + mb][n] = fast_sigmoid(d[r]);
        }
    } else if (tid < 48) {
        // Wave 1 (low half): Wf[u] = sigmoid((u - Ws) * (We - u))
        const float u = (float)(tid - 32);
        Wfs[tid - 32] = fast_sigmoid((u - Ws) * (We - u));
    }
    __syncthreads();

    // Per-thread batch b = tid. xt[b,p,bin,t] lives at x[((b*64+p)*16 + t)*10 + bin].
    const float* xrow = x + (size_t)(tid * NPATCH + p) * ROWLEN;

    float av[BIN_T];
    #pragma unroll
    for (int v = 0; v < BIN_T; ++v)
        av[v] = 1.0f - xrow[v * BIN1 + k];       // (1 - x_other[...,v])

    float acc = 0.0f;                            // sum_u log(1 - x_until*Wf)
    #pragma unroll
    for (int u = 0; u < BIN_T; ++u) {
        // Vector LDS reads: 4x ds_load_b128 per row (broadcast, conflict-free).
        const float4* wrow = reinterpret_cast<const float4*>(W1s[u]);
        const float4 w0 = wrow[0], w1 = wrow[1], w2 = wrow[2], w3 = wrow[3];

        // sum_v log(1 - av[v]*W1[u][v]) == log(prodA) + log(prodB)
        float prodA = 1.0f, prodB = 1.0f;
        prodA *= __builtin_fmaf(-av[0],  w0.x, 1.0f);
        prodA *= __builtin_fmaf(-av[1],  w0.y, 1.0f);
        prodA *= __builtin_fmaf(-av[2],  w0.z, 1.0f);
        prodA *= __builtin_fmaf(-av[3],  w0.w, 1.0f);
        prodA *= __builtin_fmaf(-av[4],  w1.x, 1.0f);
        prodA *= __builtin_fmaf(-av[5],  w1.y, 1.0f);
        prodA *= __builtin_fmaf(-av[6],  w1.z, 1.0f);
        prodA *= __builtin_fmaf(-av[7],  w1.w, 1.0f);
        prodB *= __builtin_fmaf(-av[8],  w2.x, 1.0f);
        prodB *= __builtin_fmaf(-av[9],  w2.y, 1.0f);
        prodB *= __builtin_fmaf(-av[10], w2.z, 1.0f);
        prodB *= __builtin_fmaf(-av[11], w2.w, 1.0f);
        prodB *= __builtin_fmaf(-av[12], w3.x, 1.0f);
        prodB *= __builtin_fmaf(-av[13], w3.y, 1.0f);
        prodB *= __builtin_fmaf(-av[14], w3.z, 1.0f);
        prodB *= __builtin_fmaf(-av[15], w3.w, 1.0f);

        const float s   = __logf(prodA) + __logf(prodB);
        const float xl2 = fast_rcp(1.0f - s);    // -1/(-1+s)
        const float xu  = xrow[u * BIN1 + j] * xl2;
        acc += __logf(__builtin_fmaf(-xu, Wfs[u], 1.0f));
    }
    // out = 1 - (-1/(-1+acc))
    out[(size_t)tid * OUT_DIM + idx] = 1.0f + fast_rcp(acc - 1.0f);
}

extern "C" void kernel_launch(void* const* d_in, const int* in_sizes, int n_in,
                              void* d_out, int out_size, void* d_ws, size_t ws_size,
                              hipStream_t stream) {
    (void)in_sizes; (void)n_in; (void)d_ws; (void)ws_size; (void)out_size;
    const float* x      = (const float*)d_in[0];
    const float* bound1 = (const float*)d_in[1];
    const float* bound2 = (const float*)d_in[2];
    float* out          = (float*)d_out;

    dim3 grid(OUT_DIM);   // 5760 blocks, one per (p, j, m)
    dim3 block(BATCH);    // 64 threads = 64 batches (2 waves)
    until_layer_kernel<<<grid, block, 0, stream>>>(x, bound1, bound2, out);
}